// HiPPOScale_65008624992665
// MI455X (gfx1250) — compile-verified
//
#include <hip/hip_runtime.h>

// HiPPO-Scale time-varying linear recurrence via chunked parallel scan.
// x[t] = A[t] @ x[t-1] + inputs[t,:,None]*B[t,None,:],  N=64, Bsz=64, L=4096.
// All matmuls use V_WMMA_F32_16X16X4_F32 (fp32 path; precision matters for
// 4096-long products, and we are memory-bound anyway).
//
// Pass 1 keeps the chunk A-product TRANSPOSED: P^T(t) = P^T(t-1) @ A_t^T, the
// same algebraic form as the state update S = S @ A_t^T, so both share the
// contiguous B-fragment loads of A_t (no strided scalar LDS reads in the loop).

#define N64    64
#define L_TOT  4096
#define CHUNKS 128
#define T_CH   (L_TOT / CHUNKS)   // 32 steps per chunk
#define LP     68                 // padded LDS row stride (floats): conflict-free column reads

typedef __attribute__((ext_vector_type(2))) float v2f;
typedef __attribute__((ext_vector_type(8))) float v8f;

struct F16v { float4 a, b, c, d; };   // 16 floats (one thread's staging slice)

__device__ __forceinline__ F16v ldg16(const float* p) {
  const float4* q = (const float4*)p;
  F16v f; f.a = q[0]; f.b = q[1]; f.c = q[2]; f.d = q[3]; return f;
}
__device__ __forceinline__ void stl16(float* p, const F16v& f) {
  float4* q = (float4*)p;
  q[0] = f.a; q[1] = f.b; q[2] = f.c; q[3] = f.d;
}

// 64x64 row-major global <-> padded LDS, 256 threads, coalesced float4s.
__device__ __forceinline__ void global_to_lds64(float* lds, const float* g) {
  int tid = threadIdx.x, row = tid >> 2, colb = (tid & 3) << 4;
  stl16(lds + row * LP + colb, ldg16(g + row * 64 + colb));
}
__device__ __forceinline__ void lds_to_global64(float* g, const float* lds) {
  int tid = threadIdx.x, row = tid >> 2, colb = (tid & 3) << 4;
  const float4* s = (const float4*)(lds + row * LP + colb);
  float4* d = (float4*)(g + row * 64 + colb);
  d[0] = s[0]; d[1] = s[1]; d[2] = s[2]; d[3] = s[3];
}
// Transposed variant: g[r][c] = lds[c][r] (coalesced global write; once per chunk).
__device__ __forceinline__ void ldsT_to_global64(float* g, const float* lds) {
  int tid = threadIdx.x, row = tid >> 2, colb = (tid & 3) << 4;
  float tmp[16];
#pragma unroll
  for (int j = 0; j < 16; ++j) tmp[j] = lds[(colb + j) * LP + row];
  float4* d = (float4*)(g + row * 64 + colb);
#pragma unroll
  for (int q = 0; q < 4; ++q)
    d[q] = make_float4(tmp[4 * q], tmp[4 * q + 1], tmp[4 * q + 2], tmp[4 * q + 3]);
}

// Two 16x16 output tiles (same mi, ni0 and ni0+1):  D += X_tile @ At^T_tile.
// X is (row=M, col=K) row-major padded; At is row-major so B_op=At^T frags are
// direct row reads:  b[v] = At[16*ni + lane%16][k + v + 2*(lane/16)].
__device__ __forceinline__ void mm_SAt2(const float* __restrict__ Xlds,
                                        const float* __restrict__ Atlds,
                                        int mi, int ni0, v8f& acc0, v8f& acc1) {
  const int lane = threadIdx.x & 31, lm = lane & 15, kv = lane >> 4;
  const float* xrow  = Xlds  + (mi  * 16 + lm) * LP + 2 * kv;
  const float* brow0 = Atlds + (ni0 * 16 + lm) * LP + 2 * kv;
  const float* brow1 = brow0 + 16 * LP;
#pragma unroll
  for (int k = 0; k < 64; k += 4) {
    v2f a  = *(const v2f*)(xrow  + k);
    v2f b0 = *(const v2f*)(brow0 + k);
    v2f b1 = *(const v2f*)(brow1 + k);
    acc0 = __builtin_amdgcn_wmma_f32_16x16x4_f32(false, a, false, b0, (short)0, acc0, false, false);
    acc1 = __builtin_amdgcn_wmma_f32_16x16x4_f32(false, a, false, b1, (short)0, acc1, false, false);
  }
}

// Fused pass-1 inner loop: S-tiles and P^T-tiles share B-fragments of At^T.
// 4 WMMAs per k-step from 4 contiguous 8-byte LDS loads.
__device__ __forceinline__ void mm_dual2(const float* __restrict__ Slds,
                                         const float* __restrict__ Ptlds,
                                         const float* __restrict__ Atlds,
                                         int mi, int ni0,
                                         v8f& s0, v8f& s1, v8f& p0, v8f& p1) {
  const int lane = threadIdx.x & 31, lm = lane & 15, kv = lane >> 4;
  const float* srow  = Slds  + (mi  * 16 + lm) * LP + 2 * kv;
  const float* prow  = Ptlds + (mi  * 16 + lm) * LP + 2 * kv;
  const float* brow0 = Atlds + (ni0 * 16 + lm) * LP + 2 * kv;
  const float* brow1 = brow0 + 16 * LP;
#pragma unroll
  for (int k = 0; k < 64; k += 4) {
    v2f b0 = *(const v2f*)(brow0 + k);
    v2f b1 = *(const v2f*)(brow1 + k);
    v2f as = *(const v2f*)(srow + k);
    v2f ap = *(const v2f*)(prow + k);
    s0 = __builtin_amdgcn_wmma_f32_16x16x4_f32(false, as, false, b0, (short)0, s0, false, false);
    s1 = __builtin_amdgcn_wmma_f32_16x16x4_f32(false, as, false, b1, (short)0, s1, false, false);
    p0 = __builtin_amdgcn_wmma_f32_16x16x4_f32(false, ap, false, b0, (short)0, p0, false, false);
    p1 = __builtin_amdgcn_wmma_f32_16x16x4_f32(false, ap, false, b1, (short)0, p1, false, false);
  }
}

// Write accumulator tile back to padded LDS: d[v] -> (M = v+8*kv, N = lm).
__device__ __forceinline__ void st_acc_lds(float* lds, int mi, int ni, const v8f& acc) {
  const int lane = threadIdx.x & 31, lm = lane & 15, kv = lane >> 4;
  float* base = lds + (mi * 16 + 8 * kv) * LP + ni * 16 + lm;
#pragma unroll
  for (int v = 0; v < 8; ++v) base[v * LP] = acc[v];
}

// Accumulator init from padded LDS (pass 2: s_agg tile).
__device__ __forceinline__ v8f ld_acc_lds(const float* lds, int mi, int ni) {
  const int lane = threadIdx.x & 31, lm = lane & 15, kv = lane >> 4;
  const float* base = lds + (mi * 16 + 8 * kv) * LP + ni * 16 + lm;
  v8f r;
#pragma unroll
  for (int v = 0; v < 8; ++v) r[v] = base[v * LP];
  return r;
}

// Rank-1 input injection u[b][i] = inp[b]*bst[i], laid out as a C-fragment.
__device__ __forceinline__ v8f uinit(const float* inp_s, const float* bst_s, int mi, int ni) {
  const int lane = threadIdx.x & 31, lm = lane & 15, kv = lane >> 4;
  const float bi = bst_s[ni * 16 + lm];
  v8f r;
#pragma unroll
  for (int v = 0; v < 8; ++v) r[v] = inp_s[mi * 16 + 8 * kv + v] * bi;
  return r;
}

// ---------------- Pass 1: per-chunk local scan + A-product aggregate ----------------
__global__ __launch_bounds__(256) void hippo_pass1(const float* __restrict__ A,
                                                   const float* __restrict__ inp,
                                                   const float* __restrict__ bst,
                                                   float* __restrict__ Aagg,
                                                   float* __restrict__ Sagg) {
  __shared__ float At[64 * LP], S[64 * LP], PT[64 * LP];
  __shared__ float inp_s[64], bst_s[64];
  const int c = blockIdx.x, t0 = c * T_CH, tid = threadIdx.x;
  const int row = tid >> 2, colb = (tid & 3) << 4;
  const int w = tid >> 5, mi = w >> 1, ni0 = (2 * w) & 3;

  // S = 0, P^T = I (identity is symmetric)
#pragma unroll
  for (int j = 0; j < 16; ++j) {
    S[row * LP + colb + j]  = 0.f;
    PT[row * LP + colb + j] = (row == colb + j) ? 1.f : 0.f;
  }
  // software prefetch of step 0
  F16v nextA = ldg16(A + (size_t)t0 * 4096 + row * 64 + colb);
  float smallv = 0.f;
  if (tid < 64)       smallv = inp[t0 * 64 + tid];
  else if (tid < 128) smallv = bst[t0 * 64 + (tid - 64)];

  for (int t = 0; t < T_CH; ++t) {
    stl16(At + row * LP + colb, nextA);
    if (tid < 64)       inp_s[tid] = smallv;
    else if (tid < 128) bst_s[tid - 64] = smallv;
    const int tn = (t + 1 < T_CH) ? t + 1 : t;              // clamped prefetch
    nextA = ldg16(A + (size_t)(t0 + tn) * 4096 + row * 64 + colb);
    if (tid < 64)       smallv = inp[(t0 + tn) * 64 + tid];
    else if (tid < 128) smallv = bst[(t0 + tn) * 64 + (tid - 64)];
    __syncthreads();                                        // At/inp/bst/S/PT visible

    v8f s0 = uinit(inp_s, bst_s, mi, ni0);
    v8f s1 = uinit(inp_s, bst_s, mi, ni0 + 1);
    v8f p0 = {}, p1 = {};
    mm_dual2(S, PT, At, mi, ni0, s0, s1, p0, p1);           // S@At^T + u ; PT@At^T
    __syncthreads();                                        // all reads of S,PT,At done

    st_acc_lds(S,  mi, ni0, s0); st_acc_lds(S,  mi, ni0 + 1, s1);
    st_acc_lds(PT, mi, ni0, p0); st_acc_lds(PT, mi, ni0 + 1, p1);
  }
  __syncthreads();
  ldsT_to_global64(Aagg + (size_t)c * 4096, PT);            // A_agg = (P^T)^T, row-major
  lds_to_global64(Sagg + (size_t)c * 4096, S);
}

// ---------------- Pass 2: serial scan over chunk aggregates (1 block) ----------------
// Carries overwrite the s_agg slots: slot c is read (and its prefetch for c+1 issued)
// strictly before slot c is written, ordered by the per-iteration barriers.
__global__ __launch_bounds__(256) void hippo_pass2(const float* __restrict__ Aagg,
                                                   float* __restrict__ Sagg) {
  __shared__ float At[64 * LP], S[64 * LP], U[64 * LP];
  const int tid = threadIdx.x, row = tid >> 2, colb = (tid & 3) << 4;
  const int w = tid >> 5, mi = w >> 1, ni0 = (2 * w) & 3;

#pragma unroll
  for (int j = 0; j < 16; ++j) S[row * LP + colb + j] = 0.f;  // carry[0] = 0
  F16v nextA = ldg16(Aagg + row * 64 + colb);
  F16v nextU = ldg16(Sagg + row * 64 + colb);

  for (int c = 0; c < CHUNKS; ++c) {
    stl16(At + row * LP + colb, nextA);
    stl16(U  + row * LP + colb, nextU);
    const int cn = (c + 1 < CHUNKS) ? c + 1 : c;
    nextA = ldg16(Aagg + (size_t)cn * 4096 + row * 64 + colb);
    nextU = ldg16(Sagg + (size_t)cn * 4096 + row * 64 + colb);
    __syncthreads();

    lds_to_global64(Sagg + (size_t)c * 4096, S);            // publish carry[c]
    v8f s0 = ld_acc_lds(U, mi, ni0);
    v8f s1 = ld_acc_lds(U, mi, ni0 + 1);
    mm_SAt2(S, At, mi, ni0, s0, s1);                        // carry@Aagg^T + s_agg
    __syncthreads();

    st_acc_lds(S, mi, ni0, s0); st_acc_lds(S, mi, ni0 + 1, s1);
  }
}

// ---------------- Pass 3: replay chunks from carries, emit x[t] ----------------
__global__ __launch_bounds__(256) void hippo_pass3(const float* __restrict__ A,
                                                   const float* __restrict__ inp,
                                                   const float* __restrict__ bst,
                                                   const float* __restrict__ carries,
                                                   float* __restrict__ out) {
  __shared__ float At[64 * LP], S[64 * LP];
  __shared__ float inp_s[64], bst_s[64];
  const int c = blockIdx.x, t0 = c * T_CH, tid = threadIdx.x;
  const int row = tid >> 2, colb = (tid & 3) << 4;
  const int w = tid >> 5, mi = w >> 1, ni0 = (2 * w) & 3;

  global_to_lds64(S, carries + (size_t)c * 4096);           // S = carry[c]
  F16v nextA = ldg16(A + (size_t)t0 * 4096 + row * 64 + colb);
  float smallv = 0.f;
  if (tid < 64)       smallv = inp[t0 * 64 + tid];
  else if (tid < 128) smallv = bst[t0 * 64 + (tid - 64)];

  for (int t = 0; t < T_CH; ++t) {
    stl16(At + row * LP + colb, nextA);
    if (tid < 64)       inp_s[tid] = smallv;
    else if (tid < 128) bst_s[tid - 64] = smallv;
    const int tn = (t + 1 < T_CH) ? t + 1 : t;
    nextA = ldg16(A + (size_t)(t0 + tn) * 4096 + row * 64 + colb);
    if (tid < 64)       smallv = inp[(t0 + tn) * 64 + tid];
    else if (tid < 128) smallv = bst[(t0 + tn) * 64 + (tid - 64)];
    __syncthreads();

    if (t > 0)                                              // x[t-1] store overlaps compute
      lds_to_global64(out + (size_t)(t0 + t - 1) * 4096, S);
    v8f s0 = uinit(inp_s, bst_s, mi, ni0);
    v8f s1 = uinit(inp_s, bst_s, mi, ni0 + 1);
    mm_SAt2(S, At, mi, ni0, s0, s1);
    __syncthreads();

    st_acc_lds(S, mi, ni0, s0); st_acc_lds(S, mi, ni0 + 1, s1);
  }
  __syncthreads();
  lds_to_global64(out + (size_t)(t0 + T_CH - 1) * 4096, S);
}

extern "C" void kernel_launch(void* const* d_in, const int* in_sizes, int n_in,
                              void* d_out, int out_size, void* d_ws, size_t ws_size,
                              hipStream_t stream) {
  const float* inp = (const float*)d_in[0];   // (L, B)    = (4096, 64)
  const float* A   = (const float*)d_in[1];   // (L, N, N) = (4096, 64, 64)
  const float* bst = (const float*)d_in[2];   // (L, N)    = (4096, 64)
  float* out = (float*)d_out;                 // (L, B, N)

  float* Aagg = (float*)d_ws;                 // CHUNKS * 64*64 floats (2 MB)
  float* Sagg = Aagg + (size_t)CHUNKS * 4096; // s_agg, then overwritten with carries (2 MB)

  hippo_pass1<<<CHUNKS, 256, 0, stream>>>(A, inp, bst, Aagg, Sagg);
  hippo_pass2<<<1, 256, 0, stream>>>(Aagg, Sagg);
  hippo_pass3<<<CHUNKS, 256, 0, stream>>>(A, inp, bst, Sagg, out);
}